// GRURNN_6141803233980
// MI455X (gfx1250) — compile-verified
//
#include <hip/hip_runtime.h>
#include <hip/hip_bf16.h>
#include <math.h>

// ---------------------------------------------------------------------------
// GRU-RNN (L=4 bidirectional layers, H=1024, E=2048, single step) + FC head.
// Pure GEMV workload, ~714 MB fp32 weights read once -> HBM-bandwidth bound
// (~31 us floor @ 23.3 TB/s). We use V_WMMA_F32_16X16X4_F32 as the GEMV
// engine: A = 16x4 fp32 weight tile, B = x broadcast into all 16 columns.
// ---------------------------------------------------------------------------

typedef __attribute__((ext_vector_type(2))) float v2f;
typedef __attribute__((ext_vector_type(8))) float v8f;

#define L_LAYERS 4
#define H_DIM    1024
#define E_DIM    2048
#define O_DIM    50257

// GEMV over one 16-row tile using f32 WMMA.
// wp: per-lane pointer = &W[row][koff] where row = lane&15, koff = (lane>>4)*2
// xp: per-lane pointer = &x[koff]
// Result: lane 0 holds rows 0..7 in c[0..7], lane 16 holds rows 8..15.
__device__ __forceinline__ v8f wmma_gemv(const float* __restrict__ wp,
                                         const float* __restrict__ xp,
                                         int K) {
  v8f c0 = {};
  v8f c1 = {};
  for (int k = 0; k < K; k += 8) {
    v2f a0 = *(const v2f*)(wp + k);
    v2f b0 = *(const v2f*)(xp + k);
    v2f a1 = *(const v2f*)(wp + k + 4);
    v2f b1 = *(const v2f*)(xp + k + 4);
    // 8 args: (neg_a, A, neg_b, B, c_mod, C, reuse_a, reuse_b)
    c0 = __builtin_amdgcn_wmma_f32_16x16x4_f32(false, a0, false, b0,
                                               (short)0, c0, false, false);
    c1 = __builtin_amdgcn_wmma_f32_16x16x4_f32(false, a1, false, b1,
                                               (short)0, c1, false, false);
  }
  return c0 + c1;
}

__global__ void embed_kernel(const int* __restrict__ features,
                             const float* __restrict__ table,
                             float* __restrict__ x) {
  int i = blockIdx.x * blockDim.x + threadIdx.x;
  if (i < E_DIM) x[i] = table[(size_t)features[0] * E_DIM + i];
}

// One bidirectional GRU layer. Grid: 128 blocks (2 dirs x 64 j-tiles of 16).
// Block: 192 threads = 6 waves; wave w handles (src = w/3 in {ih,hh},
// gate g = w%3 in {r,z,n}) for the same 16 rows, then 16 threads fuse gates.
__global__ void gru_layer(const float* __restrict__ x,        // (E)
                          const float* __restrict__ hidden,   // (2L, H) original
                          const float* __restrict__ w_ih,     // (L,2,3H,E)
                          const float* __restrict__ w_hh,     // (L,2,3H,H)
                          const float* __restrict__ b_ih,     // (L,2,3H)
                          const float* __restrict__ b_hh,     // (L,2,3H)
                          float* __restrict__ x_next,         // (E)
                          float* __restrict__ h_out,          // (2L, H)
                          int l) {
  __shared__ float acc[6][16];

  const int lane = threadIdx.x & 31;
  const int w    = threadIdx.x >> 5;       // 0..5
  const int d    = blockIdx.x >> 6;        // direction
  const int jb   = (blockIdx.x & 63) << 4; // row tile base within H

  const int src = w / 3;                   // 0 = ih, 1 = hh
  const int g   = w - 3 * src;             // gate: 0=r, 1=z, 2=n

  const float* h_prev = hidden + (size_t)(2 * l + d) * H_DIM;

  const int   K   = (src == 0) ? E_DIM : H_DIM;
  const float* W  = (src == 0)
      ? w_ih + ((size_t)l * 2 + d) * (3 * H_DIM) * (size_t)E_DIM
      : w_hh + ((size_t)l * 2 + d) * (3 * H_DIM) * (size_t)H_DIM;
  const float* xv = (src == 0) ? x : h_prev;

  const int row  = lane & 15;
  const int koff = (lane >> 4) << 1;       // 0 or 2
  const float* wp = W + (size_t)(g * H_DIM + jb + row) * K + koff;
  const float* xp = xv + koff;

  v8f c = wmma_gemv(wp, xp, K);

  if (lane == 0 || lane == 16) {
    const int rb = (lane >> 4) << 3;       // 0 or 8
#pragma unroll
    for (int r = 0; r < 8; ++r) acc[w][rb + r] = c[r];
  }
  __syncthreads();

  if (threadIdx.x < 16) {
    const int t = threadIdx.x;
    const int j = jb + t;
    const float* bih = b_ih + ((size_t)l * 2 + d) * (3 * H_DIM);
    const float* bhh = b_hh + ((size_t)l * 2 + d) * (3 * H_DIM);

    float ir = acc[0][t] + bih[0 * H_DIM + j];
    float iz = acc[1][t] + bih[1 * H_DIM + j];
    float in = acc[2][t] + bih[2 * H_DIM + j];
    float hr = acc[3][t] + bhh[0 * H_DIM + j];
    float hz = acc[4][t] + bhh[1 * H_DIM + j];
    float hn = acc[5][t] + bhh[2 * H_DIM + j];

    float rg = 1.0f / (1.0f + __expf(-(ir + hr)));
    float zg = 1.0f / (1.0f + __expf(-(iz + hz)));
    float ng = tanhf(in + rg * hn);
    float hp = h_prev[j];
    float hn2 = (1.0f - zg) * ng + zg * hp;

    h_out[(size_t)(2 * l + d) * H_DIM + j] = hn2;
    x_next[d * H_DIM + j] = hn2;   // concat(fwd, bwd) -> next layer input
  }
}

// FC head: out = fc_w (O x 2H) @ x + fc_b. One wave per 16-row tile;
// 8 waves per block. 412 MB streamed -> the bandwidth-dominant kernel.
__global__ void fc_kernel(const float* __restrict__ x,     // (2H)
                          const float* __restrict__ fc_w,  // (O, 2H)
                          const float* __restrict__ fc_b,  // (O)
                          float* __restrict__ out,         // (O)
                          int ntiles) {
  const int lane = threadIdx.x & 31;
  const int tile = blockIdx.x * 8 + (threadIdx.x >> 5);
  if (tile >= ntiles) return;

  const int rowbase = tile << 4;
  int grow = rowbase + (lane & 15);
  if (grow >= O_DIM) grow = O_DIM - 1;     // clamp loads for the tail tile

  const int koff = (lane >> 4) << 1;
  const float* wp = fc_w + (size_t)grow * E_DIM + koff;
  const float* xp = x + koff;

  v8f c = wmma_gemv(wp, xp, E_DIM);

  if (lane == 0 || lane == 16) {
    const int rb = (lane >> 4) << 3;
#pragma unroll
    for (int r = 0; r < 8; ++r) {
      int o = rowbase + rb + r;
      if (o < O_DIM) out[o] = c[r] + fc_b[o];
    }
  }
}

extern "C" void kernel_launch(void* const* d_in, const int* in_sizes, int n_in,
                              void* d_out, int out_size, void* d_ws, size_t ws_size,
                              hipStream_t stream) {
  const int*   features = (const int*)d_in[0];
  const float* hidden   = (const float*)d_in[1];
  const float* embed    = (const float*)d_in[2];
  const float* w_ih     = (const float*)d_in[3];
  const float* w_hh     = (const float*)d_in[4];
  const float* b_ih     = (const float*)d_in[5];
  const float* b_hh     = (const float*)d_in[6];
  const float* fc_w     = (const float*)d_in[7];
  const float* fc_b     = (const float*)d_in[8];

  float* out   = (float*)d_out;        // (1, 50257) first
  float* h_out = out + O_DIM;          // then new_hidden (2L, 1, H)

  float* xbuf0 = (float*)d_ws;         // ping-pong x vectors (2 x 8 KB)
  float* xbuf1 = xbuf0 + E_DIM;
  float* bufs[2] = {xbuf0, xbuf1};

  embed_kernel<<<(E_DIM + 255) / 256, 256, 0, stream>>>(features, embed, xbuf0);

  for (int l = 0; l < L_LAYERS; ++l) {
    gru_layer<<<128, 192, 0, stream>>>(bufs[l & 1], hidden, w_ih, w_hh,
                                       b_ih, b_hh, bufs[(l + 1) & 1], h_out, l);
  }

  const int ntiles = (O_DIM + 15) / 16;            // 3142
  fc_kernel<<<(ntiles + 7) / 8, 256, 0, stream>>>(bufs[L_LAYERS & 1], fc_w,
                                                  fc_b, out, ntiles);
}